// DropBlockForP_78743930405190
// MI455X (gfx1250) — compile-verified
//
#include <hip/hip_runtime.h>

// Problem constants (fixed by the reference)
#define BB 64
#define CC 128
#define HO 56
#define WO 56
#define HI 52              // H - bs + 1
#define WI 52
#define PLANES (BB*CC)             // 8192
#define PLANE_IN (HI*WI)           // 2704 floats = 10816 B (16B-divisible)
#define PLANE_OUT (HO*WO)          // 3136 = 98 * 32
#define WORDS_PER_PLANE (PLANE_OUT/32)   // 98
#define COUNT_M ((unsigned)PLANES*(unsigned)PLANE_OUT)  // 25690112
#define MASK_WS_OFFSET 256

// ---------------------------------------------------------------------------
// Kernel 1: init. ws is poisoned by the harness and never re-poisoned, so we
// must zero the counter ourselves every call. Also compute gamma (f64, matches
// Python-float math) from the device-resident num_batches_tracked.
// ---------------------------------------------------------------------------
__global__ void db_init(const int* __restrict__ nbt,
                        unsigned* __restrict__ dil_cnt,
                        float* __restrict__ gamma_out) {
    double n = (double)nbt[0];
    double keep = 1.0 - (0.5 / 20000.0) * n;
    if (keep < 0.5) keep = 0.5;
    double g = (1.0 - keep) / 25.0 * ((double)(WO*WO)) / ((double)(WI*WI));
    *gamma_out = (float)g;
    *dil_cnt = 0u;
}

// ---------------------------------------------------------------------------
// Kernel 2: per-plane dilation. maxpool5x5(u<gamma) == (minpool5x5(u) < gamma),
// separable. u plane is staged to LDS with CDNA5 async global->LDS b128 copies
// (ASYNCcnt), then row-min, then column-min + wave32 ballot bit-pack (32
// consecutive outputs per wave == one 32-bit mask word). Dilated bits are
// counted with integer atomics (deterministic).
// ---------------------------------------------------------------------------
__global__ __launch_bounds__(128) void db_mask(
    const float* __restrict__ u,
    unsigned* __restrict__ maskbits,
    unsigned* __restrict__ dil_cnt,
    const float* __restrict__ gamma_p)
{
    __shared__ __align__(16) float us[PLANE_IN];   // 10816 B
    __shared__ float hmin[HI * WO];                // 11648 B
    __shared__ unsigned cnt_s;

    const int tid   = threadIdx.x;
    const int plane = blockIdx.x;
    if (tid == 0) cnt_s = 0u;
    const float gamma = gamma_p[0];

    // ---- async global -> LDS: 676 x 16B chunks cover the 52x52 plane exactly
    const float* up = u + (size_t)plane * PLANE_IN;
    for (int i = tid; i < PLANE_IN / 4; i += 128) {
        // low 32 bits of the generic pointer are the LDS byte offset
        unsigned lds = (unsigned)(unsigned long long)(const void*)(us + i * 4);
        const float* g = up + i * 4;
        asm volatile("global_load_async_to_lds_b128 %0, %1, off"
                     :: "v"(lds), "v"(g) : "memory");
    }
    asm volatile("s_wait_asynccnt 0" ::: "memory");
    __syncthreads();

    // ---- horizontal 5-tap min (clipped window == +inf padding): 52 x 56
    for (int i = tid; i < HI * WO; i += 128) {
        int r = i / WO;
        int w = i - r * WO;
        int x0 = w - 4; if (x0 < 0) x0 = 0;
        int x1 = w;     if (x1 > WI - 1) x1 = WI - 1;
        const float* row = us + r * WI;
        float m = row[x0];
        #pragma unroll
        for (int x = 1; x < 5; ++x) {
            int xx = x0 + x;
            if (xx <= x1) m = fminf(m, row[xx]);
        }
        hmin[i] = m;
    }
    __syncthreads();

    // ---- vertical 5-tap min + ballot pack: wave w, iteration k handles the
    // 32 consecutive outputs [k*128 + w*32, +31] == mask word k*4 + w.
    const int lane = tid & 31;
    const int wv   = tid >> 5;
    unsigned mycnt = 0;
    for (int base = wv * 32; base < PLANE_OUT; base += 128) {
        int idx = base + lane;
        int h = idx / WO;
        int w = idx - h * WO;
        int y0 = h - 4; if (y0 < 0) y0 = 0;
        int y1 = h;     if (y1 > HI - 1) y1 = HI - 1;
        float m = hmin[y0 * WO + w];
        #pragma unroll
        for (int y = 1; y < 5; ++y) {
            int yy = y0 + y;
            if (yy <= y1) m = fminf(m, hmin[yy * WO + w]);
        }
        unsigned bits = __builtin_amdgcn_ballot_w32(m < gamma);  // dilated bits
        if (lane == 0) {
            maskbits[(size_t)plane * WORDS_PER_PLANE + (base >> 5)] = bits;
            mycnt += (unsigned)__popc(bits);
        }
    }
    if (lane == 0 && mycnt) atomicAdd(&cnt_s, mycnt);
    __syncthreads();
    if (tid == 0 && cnt_s) atomicAdd(dil_cnt, cnt_s);
}

// ---------------------------------------------------------------------------
// Kernel 3: scale = countM / count_ones  (count_ones = countM - dilated)
// ---------------------------------------------------------------------------
__global__ void db_scale(const unsigned* __restrict__ dil_cnt,
                         float* __restrict__ scale_out) {
    unsigned ones = COUNT_M - *dil_cnt;
    *scale_out = (float)COUNT_M / (float)ones;
}

// ---------------------------------------------------------------------------
// Kernel 4: out = x * (bit ? 0 : scale), float4-vectorized streaming pass.
// Each mask word feeds 8 consecutive float4s (L0/L2 reuse).
// ---------------------------------------------------------------------------
__global__ __launch_bounds__(256) void db_apply(
    const float4* __restrict__ x,
    const unsigned* __restrict__ maskbits,
    const float* __restrict__ scale_p,
    float4* __restrict__ out)
{
    int i = blockIdx.x * 256 + threadIdx.x;      // grid sized exactly
    const float s = scale_p[0];
    float4 v = x[i];
    unsigned word = maskbits[i >> 3];
    unsigned b = (unsigned)(i & 7) << 2;
    v.x = ((word >> (b + 0)) & 1u) ? 0.0f : v.x * s;
    v.y = ((word >> (b + 1)) & 1u) ? 0.0f : v.y * s;
    v.z = ((word >> (b + 2)) & 1u) ? 0.0f : v.z * s;
    v.w = ((word >> (b + 3)) & 1u) ? 0.0f : v.w * s;
    out[i] = v;
}

// ---------------------------------------------------------------------------
extern "C" void kernel_launch(void* const* d_in, const int* in_sizes, int n_in,
                              void* d_out, int out_size, void* d_ws, size_t ws_size,
                              hipStream_t stream) {
    (void)in_sizes; (void)n_in; (void)out_size; (void)ws_size;

    const float* x   = (const float*)d_in[0];
    const float* u   = (const float*)d_in[1];
    const int*   nbt = (const int*)d_in[2];
    float* out = (float*)d_out;

    unsigned char* ws = (unsigned char*)d_ws;
    unsigned* cnt      = (unsigned*)(ws + 0);
    float*    gamma    = (float*)(ws + 4);
    float*    scale    = (float*)(ws + 8);
    unsigned* maskbits = (unsigned*)(ws + MASK_WS_OFFSET);  // 3,211,264 B

    db_init <<<1, 1, 0, stream>>>(nbt, cnt, gamma);
    db_mask <<<PLANES, 128, 0, stream>>>(u, maskbits, cnt, gamma);
    db_scale<<<1, 1, 0, stream>>>(cnt, scale);
    db_apply<<<(COUNT_M / 4) / 256, 256, 0, stream>>>(
        (const float4*)x, maskbits, scale, (float4*)out);
}